// GCNModel_1288490189273
// MI455X (gfx1250) — compile-verified
//
#include <hip/hip_runtime.h>
#include <stdint.h>

// GCN 4-hop aggregation on MI455X (gfx1250, wave32).
//
// out layout (row-major, 160 cols/node):
//   cols [0,32)        = z
//   cols [32k, 32k+32) = (A^k z) / 12^k   for k = 1..4
//
// Pipeline:
//   1) init:       out[:,0:32] = z, out[:,32:160] = 0
//   2) 4x scatter: out[dst, 32k+f] += out[src, 32(k-1)+f]   (raw powers A^k z)
//   3) finalize:   out[:, 32k+f] /= 12^k                    (exact f32 constants)
//
// Mapping: one edge per wave32 -> lane f handles feature f. Gather and atomic
// per wave each touch one contiguous 128B line (perfectly coalesced). Entire
// working set (~102 MB) is resident in the 192 MB L2; bound is L2 atomic rate.

#define FEAT 32
#define COLS 160
#define EPB  256   // edges staged per block (256 threads = 8 waves, 32 edges/wave)

// ---------------------------------------------------------------- scatter
// One conv stage: for each edge e, wave-per-edge, lane-per-feature:
//   out[dst[e]*160 + col_out + lane] += out[src[e]*160 + col_in + lane]
__global__ __launch_bounds__(256) void scatter_kernel(float* __restrict__ out,
                                                      const int* __restrict__ src,
                                                      const int* __restrict__ dst,
                                                      int n_edges,
                                                      int col_in, int col_out) {
  // Interleaved (src,dst) so the hot loop reads one edge with a single
  // ds_load_b64 instead of two b32 reads.
  __shared__ int2 s_edge[EPB];

  const int t = threadIdx.x;
  const int base = blockIdx.x * EPB;

  // Stage this block's edge indices into LDS via the CDNA5 async-to-LDS path
  // (tracked with ASYNCcnt), drained to zero before the barrier.
#if defined(__gfx1250__) && __has_builtin(__builtin_amdgcn_global_load_async_to_lds_b32)
  if (base + t < n_edges) {
    __builtin_amdgcn_global_load_async_to_lds_b32(
        (__attribute__((address_space(1))) int*)(src + base + t),
        (__attribute__((address_space(3))) int*)&s_edge[t].x,
        /*offset=*/0, /*cpol=*/0);
    __builtin_amdgcn_global_load_async_to_lds_b32(
        (__attribute__((address_space(1))) int*)(dst + base + t),
        (__attribute__((address_space(3))) int*)&s_edge[t].y,
        /*offset=*/0, /*cpol=*/0);
  }
#if __has_builtin(__builtin_amdgcn_s_wait_asynccnt)
  __builtin_amdgcn_s_wait_asynccnt(0);
#else
  asm volatile("s_wait_asynccnt 0" ::: "memory");
#endif
#else
  // Fallback: plain coalesced loads into LDS.
  if (base + t < n_edges) {
    s_edge[t].x = src[base + t];
    s_edge[t].y = dst[base + t];
  }
#endif
  __syncthreads();

  const int lane = t & 31;   // feature index
  const int wave = t >> 5;   // 0..7

  // Each wave walks 32 edges, two per iteration: both gathers are issued
  // before either atomic so two 128B L2 loads are always in flight (the
  // compiler cannot hoist loads across atomics itself: it can't prove the
  // col_in read region and col_out atomic region don't alias).
  for (int i = 0; i < EPB / 8; i += 2) {
    const int eL0 = i * 8 + wave;          // uniform within the wave
    const int eL1 = eL0 + 8;
    const bool p0 = (base + eL0) < n_edges;
    const bool p1 = (base + eL1) < n_edges;

    int2 ed0 = make_int2(0, 0), ed1 = make_int2(0, 0);
    if (p0) ed0 = s_edge[eL0];             // ds_load_b64
    if (p1) ed1 = s_edge[eL1];

    float v0 = 0.0f, v1 = 0.0f;
    if (p0) v0 = out[(size_t)ed0.x * COLS + col_in + lane];
    if (p1) v1 = out[(size_t)ed1.x * COLS + col_in + lane];

    if (p0) atomicAdd(&out[(size_t)ed0.y * COLS + col_out + lane], v0);
    if (p1) atomicAdd(&out[(size_t)ed1.y * COLS + col_out + lane], v1);
  }
}

// ---------------------------------------------------------------- init
__global__ __launch_bounds__(160) void init_kernel(const float* __restrict__ z,
                                                   float* __restrict__ out) {
  const int row = blockIdx.x;
  const int c = threadIdx.x;                 // 0..159
  float v = 0.0f;
  if (c < FEAT) v = z[row * FEAT + c];
  out[(size_t)row * COLS + c] = v;
}

// ---------------------------------------------------------------- finalize
// Divide column block k by 12^k (all constants exact in f32 -> one rounding,
// matching the reference's single division per stage).
__global__ __launch_bounds__(128) void finalize_kernel(float* __restrict__ out) {
  const int row = blockIdx.x;
  const int c = 32 + threadIdx.x;          // cols 32..159
  const int k = c >> 5;                    // 1..4
  float dnm;
  if (k == 1)      dnm = 12.0f;
  else if (k == 2) dnm = 144.0f;
  else if (k == 3) dnm = 1728.0f;
  else             dnm = 20736.0f;
  const size_t i = (size_t)row * COLS + c;
  out[i] = out[i] / dnm;
}

// ---------------------------------------------------------------- launch
extern "C" void kernel_launch(void* const* d_in, const int* in_sizes, int n_in,
                              void* d_out, int out_size, void* d_ws, size_t ws_size,
                              hipStream_t stream) {
  const float* z  = (const float*)d_in[0];       // (100000, 32) f32
  const int*   ei = (const int*)d_in[1];         // (2, 1600000) indices
  const int n_nodes = in_sizes[0] / FEAT;        // 100000
  const int n_edges = in_sizes[1] / 2;           // 1600000
  const int* src = ei;                           // edge_index[0]
  const int* dst = ei + n_edges;                 // edge_index[1]
  float* out = (float*)d_out;                    // (100000, 160) f32

  // 1) out[:,0:32] = z ; out[:,32:) = 0  (also clears the 0xAA poison)
  init_kernel<<<n_nodes, COLS, 0, stream>>>(z, out);

  // 2) four cascaded scatter-add convolutions (stream-ordered dependencies)
  const int blocks = (n_edges + EPB - 1) / EPB;  // 6250
  for (int k = 1; k <= 4; ++k) {
    scatter_kernel<<<blocks, 256, 0, stream>>>(out, src, dst, n_edges,
                                               (k - 1) * FEAT, k * FEAT);
  }

  // 3) scale column blocks by 12^-k
  finalize_kernel<<<n_nodes, 128, 0, stream>>>(out);
}